// OnlineLabelSmoothing_62362925138763
// MI455X (gfx1250) — compile-verified
//
#include <hip/hip_runtime.h>
#include <stdint.h>
#include <math.h>

// OnlineLabelSmoothing for MI455X (gfx1250, wave32).
// d_out layout: [0]=loss, [1 .. K*K]=S_t_new, [1+K*K .. 1+K*K+K)=count_new

#define ALPHA_C   0.5f
#define B_SZ      4096
#define K_SZ      8192
#define NTHREADS  256
#define NWAVES    (NTHREADS / 32)
#define NCHUNK    (K_SZ / (NTHREADS * 4))   // 8 iterations of 4 floats/thread

typedef float v8f __attribute__((ext_vector_type(8)));
typedef float v2f __attribute__((ext_vector_type(2)));

// ---------------------------------------------------------------------------
// Main kernel: one block per sample row b.
//   pass 1: row max + argmax(outputs) + argmax(targets)   (row staged in LDS)
//   pass 2: e=exp(o-max) -> LDS, sumexp, dot(o, S[label]), sum(S[label])
//           -> per-row loss partial
//   pass 3: if pred==label, atomic scatter probs into S_t_new row, count+1
// ---------------------------------------------------------------------------
__global__ __launch_bounds__(NTHREADS)
void ols_main_kernel(const float* __restrict__ outputs,
                     const float* __restrict__ targets,
                     const float* __restrict__ S,
                     float* __restrict__ St_out,
                     float* __restrict__ count_out,
                     float* __restrict__ partials)
{
    __shared__ float srow[K_SZ];            // 32 KB row buffer (o, then e)
    __shared__ float r_max[NWAVES];
    __shared__ int   r_idx[NWAVES];
    __shared__ float r_tmax[NWAVES];
    __shared__ int   r_tidx[NWAVES];
    __shared__ float r_sum[NWAVES], r_p[NWAVES], r_q[NWAVES];
    __shared__ float bc_max, bc_sum;
    __shared__ int   bc_pred, bc_label;

    const int b    = blockIdx.x;
    const int t    = threadIdx.x;
    const int lane = t & 31;
    const int wv   = t >> 5;

    const float* __restrict__ orow = outputs + (size_t)b * K_SZ;
    const float* __restrict__ trow = targets + (size_t)b * K_SZ;

    // ---- CDNA5 async copy: outputs row -> LDS, 16 B per lane per op -------
    {
        unsigned lbase = (unsigned)(uintptr_t)(void*)&srow[0];
#pragma unroll
        for (int i = 0; i < NCHUNK; ++i) {
            unsigned byteoff = (unsigned)((t + i * NTHREADS) * 16);
            unsigned laddr   = lbase + byteoff;
            // GVS mode: mem = SADDR(64b) + VADDR(32b); VDST = LDS byte addr
            asm volatile("global_load_async_to_lds_b128 %0, %1, %2"
                         :: "v"(laddr), "v"(byteoff), "s"(orow)
                         : "memory");
        }
        asm volatile("s_wait_asynccnt 0" ::: "memory");
    }

    // ---------------- pass 1: max / argmax(outputs), argmax(targets) -------
    float m  = -INFINITY; int mi = 0;
    float tm = -INFINITY; int ti = 0;
#pragma unroll
    for (int i = 0; i < NCHUNK; ++i) {
        int k0 = (t + i * NTHREADS) * 4;
        float4 o  = *reinterpret_cast<const float4*>(&srow[k0]);
        float4 tg = *reinterpret_cast<const float4*>(&trow[k0]);
        if (o.x > m) { m = o.x; mi = k0;     }
        if (o.y > m) { m = o.y; mi = k0 + 1; }
        if (o.z > m) { m = o.z; mi = k0 + 2; }
        if (o.w > m) { m = o.w; mi = k0 + 3; }
        if (tg.x > tm) { tm = tg.x; ti = k0;     }
        if (tg.y > tm) { tm = tg.y; ti = k0 + 1; }
        if (tg.z > tm) { tm = tg.z; ti = k0 + 2; }
        if (tg.w > tm) { tm = tg.w; ti = k0 + 3; }
    }
    for (int off = 16; off > 0; off >>= 1) {
        float om  = __shfl_down(m, off, 32);  int omi = __shfl_down(mi, off, 32);
        if (om > m || (om == m && omi < mi)) { m = om; mi = omi; }
        float otm = __shfl_down(tm, off, 32); int oti = __shfl_down(ti, off, 32);
        if (otm > tm || (otm == tm && oti < ti)) { tm = otm; ti = oti; }
    }
    if (lane == 0) { r_max[wv] = m; r_idx[wv] = mi; r_tmax[wv] = tm; r_tidx[wv] = ti; }
    __syncthreads();
    if (t == 0) {
        float bm = r_max[0];  int bi  = r_idx[0];
        float bt = r_tmax[0]; int bti = r_tidx[0];
        for (int w = 1; w < NWAVES; ++w) {
            if (r_max[w]  > bm || (r_max[w]  == bm && r_idx[w]  < bi))  { bm = r_max[w];  bi  = r_idx[w];  }
            if (r_tmax[w] > bt || (r_tmax[w] == bt && r_tidx[w] < bti)) { bt = r_tmax[w]; bti = r_tidx[w]; }
        }
        bc_max = bm; bc_pred = bi; bc_label = bti;
    }
    __syncthreads();

    // ---------------- pass 2: exp, sumexp, dot with gathered S row ---------
    const float rm    = bc_max;
    const int   label = bc_label;
    const float* __restrict__ Srow = S + (size_t)label * K_SZ;
    float sum = 0.f, p = 0.f, q = 0.f;
#pragma unroll
    for (int i = 0; i < NCHUNK; ++i) {
        int k0 = (t + i * NTHREADS) * 4;
        float4 o  = *reinterpret_cast<const float4*>(&srow[k0]);
        float4 sv = *reinterpret_cast<const float4*>(&Srow[k0]);
        float4 e;
        e.x = __expf(o.x - rm); e.y = __expf(o.y - rm);
        e.z = __expf(o.z - rm); e.w = __expf(o.w - rm);
        *reinterpret_cast<float4*>(&srow[k0]) = e;     // overwrite o with e
        sum += (e.x + e.y) + (e.z + e.w);
        p   += o.x * sv.x + o.y * sv.y + o.z * sv.z + o.w * sv.w;
        q   += (sv.x + sv.y) + (sv.z + sv.w);
    }
    for (int off = 16; off > 0; off >>= 1) {
        sum += __shfl_down(sum, off, 32);
        p   += __shfl_down(p,   off, 32);
        q   += __shfl_down(q,   off, 32);
    }
    if (lane == 0) { r_sum[wv] = sum; r_p[wv] = p; r_q[wv] = q; }
    __syncthreads();
    if (t == 0) {
        float ts = 0.f, tp = 0.f, tq = 0.f;
        for (int w = 0; w < NWAVES; ++w) { ts += r_sum[w]; tp += r_p[w]; tq += r_q[w]; }
        bc_sum = ts;
        float lse   = __logf(ts);
        float o_lab = orow[label];                    // single global scalar
        float ce_h  = -(o_lab - rm - lse);            // -ls[b, label]
        float ce_s  = -(tp - (rm + lse) * tq);        // -(sum o*S - (m+lse)*sum S)
        partials[b] = (ALPHA_C * ce_h + (1.0f - ALPHA_C) * ce_s) * (1.0f / (float)B_SZ);
    }
    __syncthreads();

    // ---------------- pass 3: rare scatter of probs into S_t_new -----------
    if (bc_pred == bc_label) {
        float inv = 1.0f / bc_sum;
        float* __restrict__ drow = St_out + (size_t)bc_label * K_SZ;
#pragma unroll
        for (int i = 0; i < NCHUNK; ++i) {
            int k0 = (t + i * NTHREADS) * 4;
            float4 e = *reinterpret_cast<const float4*>(&srow[k0]);
            atomicAdd(&drow[k0 + 0], e.x * inv);
            atomicAdd(&drow[k0 + 1], e.y * inv);
            atomicAdd(&drow[k0 + 2], e.z * inv);
            atomicAdd(&drow[k0 + 3], e.w * inv);
        }
        if (t == 0) atomicAdd(&count_out[bc_label], 1.0f);
    }
}

// ---------------------------------------------------------------------------
// Final loss reduction: 4096 partials -> scalar, via v_wmma_f32_16x16x4_f32
// with an all-ones B matrix.  A layout (16x4 f32): lanes 0-15 supply
// A[lane][0..1] in V0/V1, lanes 16-31 supply A[lane-16][2..3].  With
// a = {s_lane, 0} and B = ones:  D[m][n] = s_m + s_{m+16} for every n.
// D layout: lane n holds D[0..7][n] (lanes 0-15) / D[8..15][n-16] (16-31),
// so total = sum of 8 accumulators on lane0 + lane16 -> shfl_xor(16).
// ---------------------------------------------------------------------------
__global__ __launch_bounds__(32)
void ols_loss_reduce_kernel(const float* __restrict__ partials,
                            float* __restrict__ loss_out)
{
    const int lane = threadIdx.x;
    float s = 0.f;
    for (int i = lane; i < B_SZ; i += 32) s += partials[i];   // fixed order

#if __has_builtin(__builtin_amdgcn_wmma_f32_16x16x4_f32)
    v2f a;  a.x = s;   a.y = 0.f;
    v2f bo; bo.x = 1.f; bo.y = 1.f;
    v8f c = {};
    c = __builtin_amdgcn_wmma_f32_16x16x4_f32(
            /*neg_a=*/false, a, /*neg_b=*/false, bo,
            /*c_mod=*/(short)0, c, /*reuse_a=*/false, /*reuse_b=*/false);
    float r = ((c[0] + c[1]) + (c[2] + c[3])) + ((c[4] + c[5]) + (c[6] + c[7]));
    r += __shfl_xor(r, 16, 32);     // rows 0-7 + rows 8-15
    if (lane == 0) loss_out[0] = r;
#else
    for (int off = 16; off > 0; off >>= 1) s += __shfl_down(s, off, 32);
    if (lane == 0) loss_out[0] = s;
#endif
}

// ---------------------------------------------------------------------------
extern "C" void kernel_launch(void* const* d_in, const int* in_sizes, int n_in,
                              void* d_out, int out_size, void* d_ws, size_t ws_size,
                              hipStream_t stream)
{
    const float* outputs = (const float*)d_in[0];   // [B, K]
    const float* targets = (const float*)d_in[1];   // [B, K]
    const float* S       = (const float*)d_in[2];   // [K, K]
    const float* S_t     = (const float*)d_in[3];   // [K, K]
    const float* count   = (const float*)d_in[4];   // [K, 1]

    float* out       = (float*)d_out;
    float* loss_out  = out;                                  // [1]
    float* St_out    = out + 1;                              // [K*K]
    float* count_out = out + 1 + (size_t)K_SZ * K_SZ;        // [K]
    float* partials  = (float*)d_ws;                         // [B]

    // Carry state to output first (stream-ordered before the scatter kernel).
    hipMemcpyAsync(St_out, S_t, (size_t)K_SZ * K_SZ * sizeof(float),
                   hipMemcpyDeviceToDevice, stream);
    hipMemcpyAsync(count_out, count, (size_t)K_SZ * sizeof(float),
                   hipMemcpyDeviceToDevice, stream);

    ols_main_kernel<<<B_SZ, NTHREADS, 0, stream>>>(
        outputs, targets, S, St_out, count_out, partials);

    ols_loss_reduce_kernel<<<1, 32, 0, stream>>>(partials, loss_out);
}